// GridPool_29351806501462
// MI455X (gfx1250) — compile-verified
//
#include <hip/hip_runtime.h>

typedef float v2f __attribute__((ext_vector_type(2)));
typedef float v8f __attribute__((ext_vector_type(8)));

#define NPTS    300000
#define NTILE16 18750          // NPTS / 16 (exact)
#define CIN     64
#define COUT    128
#define NBINS   (4*32*32*32)   // 131072 compact voxel bins (order-isomorphic to ref key)
#define LDSPAD  68             // 64 + 4 pad -> conflict-free ds_load_b64
#define EPSBN   1e-5f

// ---------------------------------------------------------------------------
// Init: zero outputs / histogram / stat accumulators, fill batch_out with
// int32-max (empty segment_min identity), start-mins with +FLT_MAX bits.
// ---------------------------------------------------------------------------
__global__ void k_init(float* __restrict__ out, int* __restrict__ bins,
                       float* __restrict__ stats, int* __restrict__ startBits) {
  long long i = (long long)blockIdx.x * blockDim.x + threadIdx.x;
  const long long zeroEnd = 133LL * NPTS;   // coord(3N)+feat(128N)+cluster(N)+counts(N)
  const long long outEnd  = 134LL * NPTS;   // + batch(N)
  if (i < zeroEnd) { out[i] = 0.0f; return; }
  if (i < outEnd)  { out[i] = 2147483648.0f; return; }   // int32 max as f32
  long long j = i - outEnd;
  if (j < NBINS)   { bins[j] = 0; return; }
  j -= NBINS;
  if (j < 256)     { stats[j] = 0.0f; return; }          // sum[128], sumsq[128]
  j -= 256;
  if (j < 12)      { startBits[j] = 0x7F7FFFFF; return; } // +FLT_MAX bits
}

// ---------------------------------------------------------------------------
// batch id (searchsorted right) + per-scene min coordinate (int-bit atomicMin,
// valid because coords are non-negative).
// ---------------------------------------------------------------------------
__global__ void k_batch_start(const float* __restrict__ coord,
                              const int* __restrict__ offset,
                              int* __restrict__ batchI,
                              int* __restrict__ startBits) {
  int i = blockIdx.x * blockDim.x + threadIdx.x;
  if (i >= NPTS) return;
  int b = (i >= offset[0]) + (i >= offset[1]) + (i >= offset[2]);
  batchI[i] = b;
  #pragma unroll
  for (int d = 0; d < 3; ++d)
    atomicMin(&startBits[b * 3 + d], __float_as_int(coord[3 * i + d]));
}

// ---------------------------------------------------------------------------
// Compact voxel key + histogram.
// ---------------------------------------------------------------------------
__global__ void k_vkey(const float* __restrict__ coord,
                       const int* __restrict__ batchI,
                       const int* __restrict__ startBits,
                       int* __restrict__ kbArr, int* __restrict__ bins) {
  int i = blockIdx.x * blockDim.x + threadIdx.x;
  if (i >= NPTS) return;
  int b = batchI[i];
  int vx = (int)floorf((coord[3*i+0] - __int_as_float(startBits[b*3+0])) * 2.0f);
  int vy = (int)floorf((coord[3*i+1] - __int_as_float(startBits[b*3+1])) * 2.0f);
  int vz = (int)floorf((coord[3*i+2] - __int_as_float(startBits[b*3+2])) * 2.0f);
  int kb = ((b * 32 + vx) * 32 + vy) * 32 + vz;   // < NBINS, order-preserving
  kbArr[i] = kb;
  atomicAdd(&bins[kb], 1);
}

// ---------------------------------------------------------------------------
// Single-workgroup rank scan over the 131072-bin histogram:
// rank[bin] = #nonempty bins before it  == jnp.unique sorted cluster id.
// Also emits counts / batch per cluster.
// ---------------------------------------------------------------------------
__global__ __launch_bounds__(1024) void k_scan(const int* __restrict__ bins,
                                               int* __restrict__ rank,
                                               float* __restrict__ countsOut,
                                               float* __restrict__ batchOut) {
  __shared__ int lds[1024];
  const int CH = NBINS / 1024;           // 128 bins per thread
  int tid = threadIdx.x;
  int base = tid * CH;
  int cnt = 0;
  for (int i = 0; i < CH; ++i) cnt += (bins[base + i] > 0);
  lds[tid] = cnt;
  __syncthreads();
  for (int off = 1; off < 1024; off <<= 1) {   // Hillis-Steele inclusive scan
    int t = (tid >= off) ? lds[tid - off] : 0;
    __syncthreads();
    lds[tid] += t;
    __syncthreads();
  }
  int r = lds[tid] - cnt;                // exclusive prefix
  for (int i = 0; i < CH; ++i) {
    int b = base + i, c = bins[b];
    if (c > 0) {
      rank[b] = r;
      countsOut[r] = (float)c;
      batchOut[r] = (float)(b >> 15);    // scene id = kb / 32768
      ++r;
    }
  }
}

// ---------------------------------------------------------------------------
// cluster id per point + segment-sum of coords.
// ---------------------------------------------------------------------------
__global__ void k_assign(const float* __restrict__ coord,
                         const int* __restrict__ kbArr,
                         const int* __restrict__ rank,
                         int* __restrict__ clusterI,
                         float* __restrict__ clusterOut,
                         float* __restrict__ coordOut) {
  int i = blockIdx.x * blockDim.x + threadIdx.x;
  if (i >= NPTS) return;
  int c = rank[kbArr[i]];
  clusterI[i] = c;
  clusterOut[i] = (float)c;
  #pragma unroll
  for (int d = 0; d < 3; ++d)
    atomicAdd(&coordOut[(long long)c * 3 + d], coord[3 * i + d]);
}

__global__ void k_coord_div(const float* __restrict__ countsOut,
                            float* __restrict__ coordOut) {
  int i = blockIdx.x * blockDim.x + threadIdx.x;
  if (i >= NPTS) return;
  float cnt = fmaxf(countsOut[i], 1.0f);
  #pragma unroll
  for (int d = 0; d < 3; ++d)
    coordOut[3 * i + d] = coordOut[3 * i + d] / cnt;
}

// ---------------------------------------------------------------------------
// WMMA f32 16x16x4 GEMM core. Each wave: one 16-row tile x full 128 columns.
// W staged transposed in LDS ([128][68] padded) so B fragments are contiguous
// 8-byte ds loads, bank-conflict free.
// ---------------------------------------------------------------------------
__device__ __forceinline__ void gemm_tile(const float* __restrict__ feat,
                                          const float* __restrict__ ldsW,
                                          int tile16, int lane, v8f acc[8]) {
  int row   = tile16 * 16 + (lane & 15);
  int khalf = (lane >> 4) * 2;                       // half-wave K-pair select
  const float* arow = feat + (long long)row * CIN + khalf;
  for (int k0 = 0; k0 < CIN; k0 += 4) {
    v2f a = *(const v2f*)(arow + k0);                // A: rows x K{k0..k0+3}
    #pragma unroll
    for (int t = 0; t < 8; ++t) {
      const float* bp = ldsW + (t * 16 + (lane & 15)) * LDSPAD + k0 + khalf;
      v2f b = *(const v2f*)bp;                       // B: Wt[j][k], Wt[j][k+1]
      acc[t] = __builtin_amdgcn_wmma_f32_16x16x4_f32(
          false, a, false, b, (short)0, acc[t], false, false);
    }
  }
}

__device__ __forceinline__ void load_W_lds(const float* __restrict__ W,
                                           float* __restrict__ ldsW, int tid) {
  for (int idx = tid; idx < CIN * COUT; idx += 256) {
    int k = idx >> 7, j = idx & 127;                 // coalesced global reads
    ldsW[j * LDSPAD + k] = W[k * COUT + j];
  }
}

// Pass 1: per-channel sum / sumsq of x = feat @ W (LDS reduce, then global).
__global__ __launch_bounds__(256) void k_gemm_stats(const float* __restrict__ feat,
                                                    const float* __restrict__ W,
                                                    float* __restrict__ sumC,
                                                    float* __restrict__ sqC) {
  __shared__ float ldsW[COUT * LDSPAD];
  __shared__ float lsum[COUT], lsq[COUT];
  int tid = threadIdx.x;
  load_W_lds(W, ldsW, tid);
  if (tid < COUT) { lsum[tid] = 0.0f; lsq[tid] = 0.0f; }
  __syncthreads();
  int wave = tid >> 5, lane = tid & 31;
  int tile16 = blockIdx.x * 8 + wave;                // wave-uniform guard
  if (tile16 < NTILE16) {
    v8f acc[8] = {};
    gemm_tile(feat, ldsW, tile16, lane, acc);
    #pragma unroll
    for (int t = 0; t < 8; ++t) {
      float s = 0.0f, q = 0.0f;
      #pragma unroll
      for (int v = 0; v < 8; ++v) { float x = acc[t][v]; s += x; q += x * x; }
      int j = t * 16 + (lane & 15);
      atomicAdd(&lsum[j], s);                        // ds_add_f32
      atomicAdd(&lsq[j], q);
    }
  }
  __syncthreads();
  if (tid < COUT) {
    atomicAdd(&sumC[tid], lsum[tid]);
    atomicAdd(&sqC[tid], lsq[tid]);
  }
}

__global__ void k_stats_final(const float* __restrict__ sumC,
                              const float* __restrict__ sqC,
                              const float* __restrict__ gamma,
                              float* __restrict__ meanC,
                              float* __restrict__ scaleC) {
  int j = threadIdx.x;
  if (j >= COUT) return;
  float inv_n = 1.0f / (float)NPTS;
  float m = sumC[j] * inv_n;
  float var = sqC[j] * inv_n - m * m;                // biased variance
  meanC[j] = m;
  scaleC[j] = gamma[j] * rsqrtf(var + EPSBN);
}

// Pass 2: recompute x, apply BN+ReLU, segment-max scatter (int-bit atomicMax,
// valid because ReLU output >= 0 and feat_out is pre-zeroed).
__global__ __launch_bounds__(256) void k_gemm_scatter(const float* __restrict__ feat,
                                                      const float* __restrict__ W,
                                                      const float* __restrict__ meanC,
                                                      const float* __restrict__ scaleC,
                                                      const float* __restrict__ beta,
                                                      const int* __restrict__ clusterI,
                                                      float* __restrict__ featOut) {
  __shared__ float ldsW[COUT * LDSPAD];
  int tid = threadIdx.x;
  load_W_lds(W, ldsW, tid);
  __syncthreads();
  int wave = tid >> 5, lane = tid & 31;
  int tile16 = blockIdx.x * 8 + wave;
  if (tile16 >= NTILE16) return;
  v8f acc[8] = {};
  gemm_tile(feat, ldsW, tile16, lane, acc);
  // C layout: row = tile*16 + v + 8*(lane>=16), col = t*16 + (lane&15)
  int carr[8];
  #pragma unroll
  for (int v = 0; v < 8; ++v)
    carr[v] = clusterI[tile16 * 16 + v + 8 * (lane >> 4)];
  #pragma unroll
  for (int t = 0; t < 8; ++t) {
    int j = t * 16 + (lane & 15);
    float m = meanC[j], sc = scaleC[j], bt = beta[j];
    #pragma unroll
    for (int v = 0; v < 8; ++v) {
      float y = fmaxf(sc * (acc[t][v] - m) + bt, 0.0f);
      atomicMax((int*)&featOut[(long long)carr[v] * COUT + j], __float_as_int(y));
    }
  }
}

// ---------------------------------------------------------------------------
extern "C" void kernel_launch(void* const* d_in, const int* in_sizes, int n_in,
                              void* d_out, int out_size, void* d_ws, size_t ws_size,
                              hipStream_t stream) {
  const float* coord  = (const float*)d_in[0];
  const float* feat   = (const float*)d_in[1];
  const int*   offset = (const int*)d_in[2];
  const float* W      = (const float*)d_in[3];
  const float* gamma  = (const float*)d_in[4];
  const float* beta   = (const float*)d_in[5];

  float* out        = (float*)d_out;
  float* coordOut   = out;                  // [N,3]
  float* featOut    = out + 3LL * NPTS;     // [N,128]
  float* clusterOut = out + 131LL * NPTS;   // [N]
  float* countsOut  = out + 132LL * NPTS;   // [N]
  float* batchOut   = out + 133LL * NPTS;   // [N]

  int* ws        = (int*)d_ws;
  int* bins      = ws;                      // [NBINS]
  int* rank      = ws + NBINS;              // [NBINS]
  int* kbArr     = ws + 2 * NBINS;          // [N]
  int* clusterI  = kbArr + NPTS;            // [N]
  int* batchI    = clusterI + NPTS;         // [N]
  int* startBits = batchI + NPTS;           // [12]
  float* stats   = (float*)(startBits + 12);
  float* sumC = stats, *sqC = stats + 128, *meanC = stats + 256, *scaleC = stats + 384;

  long long tot = 134LL * NPTS + NBINS + 256 + 12;
  int initBlocks = (int)((tot + 255) / 256);
  int nb = (NPTS + 255) / 256;
  int gb = (NTILE16 + 7) / 8;

  k_init<<<initBlocks, 256, 0, stream>>>(out, bins, stats, startBits);
  k_batch_start<<<nb, 256, 0, stream>>>(coord, offset, batchI, startBits);
  k_vkey<<<nb, 256, 0, stream>>>(coord, batchI, startBits, kbArr, bins);
  k_scan<<<1, 1024, 0, stream>>>(bins, rank, countsOut, batchOut);
  k_assign<<<nb, 256, 0, stream>>>(coord, kbArr, rank, clusterI, clusterOut, coordOut);
  k_coord_div<<<nb, 256, 0, stream>>>(countsOut, coordOut);
  k_gemm_stats<<<gb, 256, 0, stream>>>(feat, W, sumC, sqC);
  k_stats_final<<<1, 128, 0, stream>>>(sumC, sqC, gamma, meanC, scaleC);
  k_gemm_scatter<<<gb, 256, 0, stream>>>(feat, W, meanC, scaleC, beta, clusterI, featOut);
}